// Sabr_46909632807652
// MI455X (gfx1250) — compile-verified
//
#include <hip/hip_runtime.h>
#include <hip/hip_bf16.h>
#include <math.h>

typedef float     v2f  __attribute__((ext_vector_type(2)));
typedef float     v8f  __attribute__((ext_vector_type(8)));

#define DT                 0.003968253968253968f   /* 1/252 */
#define EPS_VAR            0.01f
#define LOG_EPS_P         -13.815510557964274f     /* log(1e-6) */
#define NEG_HALF_LOG_2PI  -0.9189385332046727f
#define TWO_LOG2E          2.8853900817779268f     /* 2*log2(e) */

#define MU_BLOCKS    64
#define MU_THREADS   256
#define TILE         4096
#define MAIN_THREADS 1024

// Sum 64 f32 values (2 per lane of one wave32) through the CDNA5 matrix pipe.
// A = all-ones 16x4 => C[m][n] = sum_k B[k][n]; every input occupies exactly
// one (k,n) slot of B, so the sum over C row 0 (lanes 0..15 of c[0]) is the
// total regardless of B's exact lane layout. Lane 0 returns the total.
__device__ __forceinline__ float wmma_sum64(float vlo, float vhi) {
  v2f a; a[0] = 1.0f; a[1] = 1.0f;
  v2f b; b[0] = vlo;  b[1] = vhi;
  v8f c = {};
  c = __builtin_amdgcn_wmma_f32_16x16x4_f32(false, a, false, b, (short)0, c,
                                            false, false);
  float c0 = c[0];
  // C rows replicated (A all-ones); lanes 16..31 duplicate lanes 0..15.
  #pragma unroll
  for (int off = 8; off >= 1; off >>= 1) c0 += __shfl_xor(c0, off, 32);
  return c0;
}

// ---------------------------------------------------------------- mu pass 1
__global__ void sabr_mu_partials(const float* __restrict__ s,
                                 const int* __restrict__ tptr,
                                 const int* __restrict__ wptr,
                                 float* __restrict__ partials) {
  __shared__ float red[MU_THREADS];
  int t = tptr[0], w = wptr[0];
  int start = t - w; if (start < 0) start = 0;
  int N = t - start;
  float acc = 0.0f;
  for (int i = blockIdx.x * MU_THREADS + threadIdx.x; i < N;
       i += MU_BLOCKS * MU_THREADS) {
    float s0 = s[start + i];
    acc += (s[start + i + 1] - s0) / s0;
  }
  red[threadIdx.x] = acc;
  __syncthreads();
  for (int st = MU_THREADS / 2; st > 0; st >>= 1) {
    if (threadIdx.x < st) red[threadIdx.x] += red[threadIdx.x + st];
    __syncthreads();
  }
  if (threadIdx.x == 0) partials[blockIdx.x] = red[0];
}

// ---------------------------------------------------------------- mu pass 2
__global__ void sabr_mu_finalize(const int* __restrict__ tptr,
                                 const int* __restrict__ wptr,
                                 const float* __restrict__ partials,
                                 float* __restrict__ mu_out) {
  int t = tptr[0], w = wptr[0];
  int start = t - w; if (start < 0) start = 0;
  int N = t - start;
  int l = threadIdx.x;                      // exactly one wave32, EXEC all 1s
  float total = wmma_sum64(partials[l], partials[l + 32]);
  if (l == 0) mu_out[0] = (N > 0) ? total / (float)N : 0.0f;
}

// ------------------------------------------------- tiled scan + likelihood
__global__ void sabr_ll_kernel(const float* __restrict__ s,
                               const float* __restrict__ atb,
                               const float* __restrict__ lsg,
                               const float* __restrict__ atr,
                               const float* __restrict__ d0p,
                               const int* __restrict__ tptr,
                               const int* __restrict__ wptr,
                               const float* __restrict__ mup,
                               float* __restrict__ out) {
  __shared__ float shH[TILE];   // exp(c2*k + (beta-2)*log sp[k])
  __shared__ float shP[TILE];   // 2*log2(e) * rho*sigma^2*(lsd_k - mu*DT)
  __shared__ float shX[TILE];   // y_path = 2*log2(delta) for the tile

  const int tid = threadIdx.x;
  int t = tptr[0], w = wptr[0];
  int start = t - w; if (start < 0) start = 0;
  int N = t - start;
  if (N <= 0) { if (tid == 0) out[0] = 0.0f; return; }

  const float* sp = s + start;
  const float mu    = mup[0];
  const float beta  = tanhf(atb[0]) + 1.0f;
  const float sigma = expf(lsg[0]);
  const float rho   = tanhf(atr[0]);
  const float sig2   = sigma * sigma;
  const float sig2dt = sig2 * DT;
  const float st2    = (1.0f - rho * rho) * sig2dt;    // sigma_tilde^2
  const float Kc   = -0.5f * sig2dt + 0.5f * rho * sig2 * DT + 0.5f * st2;
  const float Kc2  = TWO_LOG2E * Kc;                   // log2-domain constant
  const float rs2l = TWO_LOG2E * rho * sig2;           // prescaled P factor
  const float bm2  = beta - 2.0f;
  const float c2   = (2.0f - beta) * mu * DT;
  const float cf   = expf(c2);          // shifts H_k -> scan coefficient G_k
  const float muDT = mu * DT;

  // Recurrence state in log2 domain: y = 2*log2(delta) => delta^2 = exp2(y).
  float y   = TWO_LOG2E * logf(d0p[0]); // meaningful on thread 0 only
  float acc = 0.0f;
  const int ntiles = (N + TILE - 1) / TILE;

  for (int tile = 0; tile < ntiles; ++tile) {
    const int k0 = tile * TILE;

    // Parallel per-step coefficients for this tile.
    for (int i = tid; i < TILE; i += MAIN_THREADS) {
      int k = k0 + i;
      if (k < N) {
        float sk  = sp[k];
        float lsk = logf(sk);
        shH[i] = expf(c2 * (float)k + bm2 * lsk);
        shP[i] = (k < N - 1) ? rs2l * (logf(sp[k + 1]) - lsk - muDT) : 0.0f;
      } else { shH[i] = 0.0f; shP[i] = 0.0f; }
    }
    __syncthreads();

    // Sequential recurrence (irreducible chain), thread 0 only.
    // Critical path per step: v_exp_f32 -> v_mul -> v_max -> v_rcp -> v_fma.
    //   var = max(G_j * exp2(y), EPS_VAR)   [relu(v-e)+e == max(v,e)]
    //   y  += Kc2 + P2_j * rcp(var)
    // Branch-free: shP is zero at j==N-1, and the post-final y is never read.
    if (tid == 0) {
      int lim = N - k0; if (lim > TILE) lim = TILE;
      for (int i = 0; i < lim; ++i) {
        shX[i] = y;
        float e   = __builtin_amdgcn_exp2f(y);       // delta^2 (on chain)
        float lv  = (shH[i] * cf) * e;               // G off-chain, 1 chained mul
        float var = fmaxf(lv, EPS_VAR);
        float r   = __builtin_amdgcn_rcpf(var);      // var >= 0.01: safe
        y = (y + Kc2) + shP[i] * r;
      }
    }
    __syncthreads();

    // Parallel likelihood terms for this tile.
    for (int i = tid; i < TILE; i += MAIN_THREADS) {
      int k = k0 + i;
      if (k < N) {
        float varp  = shH[i] * exp2f(shX[i]);        // unclamped, as in ref
        float vdt   = varp * DT;
        float mean  = logf(sp[k]) + (mu - 0.5f * varp) * DT;
        float xo    = logf(sp[k + 1]);               // log s_all[start+k+1]
        float z     = (xo - mean) / sqrtf(vdt);
        float lp    = -0.5f * z * z - 0.5f * logf(vdt) + NEG_HALF_LOG_2PI;
        acc += fmaxf(lp, LOG_EPS_P);  // log(max(exp(lp),eps)) == max(lp,log eps)
      }
    }
    __syncthreads();
  }

  // Block reduction: LDS tree 1024 -> 64, then WMMA 64 -> 1.
  shH[tid] = acc;
  __syncthreads();
  for (int st = MAIN_THREADS / 2; st >= 64; st >>= 1) {
    if (tid < st) shH[tid] += shH[tid + st];
    __syncthreads();
  }
  if (tid < 32) {                       // wave 0 only: EXEC all 1s for WMMA
    float total = wmma_sum64(shH[tid], shH[tid + 32]);
    if (tid == 0) out[0] = total;
  }
}

extern "C" void kernel_launch(void* const* d_in, const int* in_sizes, int n_in,
                              void* d_out, int out_size, void* d_ws,
                              size_t ws_size, hipStream_t stream) {
  (void)in_sizes; (void)n_in; (void)out_size; (void)ws_size;
  const float* s   = (const float*)d_in[0];
  const float* atb = (const float*)d_in[1];
  const float* lsg = (const float*)d_in[2];
  const float* atr = (const float*)d_in[3];
  const float* d0  = (const float*)d_in[4];
  const int*   tp  = (const int*)d_in[5];
  const int*   wp  = (const int*)d_in[6];
  float* out = (float*)d_out;
  float* ws  = (float*)d_ws;
  float* partials = ws;        // 64 floats
  float* mu       = ws + 64;   // 1 float

  hipLaunchKernelGGL(sabr_mu_partials, dim3(MU_BLOCKS), dim3(MU_THREADS), 0,
                     stream, s, tp, wp, partials);
  hipLaunchKernelGGL(sabr_mu_finalize, dim3(1), dim3(32), 0, stream,
                     tp, wp, partials, mu);
  hipLaunchKernelGGL(sabr_ll_kernel, dim3(1), dim3(MAIN_THREADS), 0, stream,
                     s, atb, lsg, atr, d0, tp, wp, mu, out);
}